// CilLayer_57707180589155
// MI455X (gfx1250) — compile-verified
//
#include <hip/hip_runtime.h>
#include <math.h>

// ---------------------------------------------------------------------------
// Elementwise transform on [B, L, 3] f32 -> [B, L, 2] f32.
//   phi = (clip(x,-1,1) + 1) * pi / 2
//   n   = sqrt(y*y + z*z)
//   psi = acos(y/n), negated where z < 0
//   out = [-phi * 180/pi, psi * 180/pi]
// Memory-bound: 168 MB total traffic; fits in MI455X's 192 MB L2 across
// graph replays, so default (RT) cache hints are used intentionally.
// 4 points/thread -> 3x global_load_b128 + 2x global_store_b128 per thread.
// ---------------------------------------------------------------------------

__device__ __forceinline__ float2 cil_point(float x, float y, float z) {
#pragma clang fp contract(off)
    const float pi_f  = 3.14159265358979323846f;   // rounds to f32 pi (matches jnp.float32(math.pi))
    const float fac_f = 57.29577951308232f;        // (float)(180.0 / pi), matches jnp.float32(FACTOR)

    float cx  = fminf(fmaxf(x, -1.0f), 1.0f);
    float phi = ((cx + 1.0f) * pi_f) * 0.5f;       // same op order as reference; *0.5 == /2 exactly

    float yy = y * y;
    float zz = z * z;
    float n  = sqrtf(yy + zz);
    float u  = y / n;                               // IEEE divide: acos is ULP-sensitive near |u|~1
    float psi = acosf(u);
    psi = (z < 0.0f) ? -psi : psi;

    float2 r;
    r.x = -phi * fac_f;
    r.y =  psi * fac_f;
    return r;
}

// Main vectorized path: thread j handles points 4j..4j+3.
// Input footprint per thread = 48 B = 3 float4; output = 32 B = 2 float4.
__global__ void __launch_bounds__(256)
cil_kernel_vec4(const float4* __restrict__ in4,
                float4* __restrict__ out4,
                int nquad) {
    int j = blockIdx.x * blockDim.x + threadIdx.x;
    if (j >= nquad) return;

    float4 a = in4[3 * j + 0];   // p0.x p0.y p0.z p1.x
    float4 b = in4[3 * j + 1];   // p1.y p1.z p2.x p2.y
    float4 c = in4[3 * j + 2];   // p2.z p3.x p3.y p3.z

    float2 r0 = cil_point(a.x, a.y, a.z);
    float2 r1 = cil_point(a.w, b.x, b.y);
    float2 r2 = cil_point(b.z, b.w, c.x);
    float2 r3 = cil_point(c.y, c.z, c.w);

    float4 o0; o0.x = r0.x; o0.y = r0.y; o0.z = r1.x; o0.w = r1.y;
    float4 o1; o1.x = r2.x; o1.y = r2.y; o1.z = r3.x; o1.w = r3.y;

    out4[2 * j + 0] = o0;
    out4[2 * j + 1] = o1;
}

// Scalar tail for point counts not divisible by 4 (not hit for 128*65536,
// but kept so kernel_launch is correct for any shape).
__global__ void __launch_bounds__(256)
cil_kernel_tail(const float* __restrict__ in,
                float* __restrict__ out,
                int start, int npts) {
    int p = start + blockIdx.x * blockDim.x + threadIdx.x;
    if (p >= npts) return;
    float x = in[3 * p + 0];
    float y = in[3 * p + 1];
    float z = in[3 * p + 2];
    float2 r = cil_point(x, y, z);
    out[2 * p + 0] = r.x;
    out[2 * p + 1] = r.y;
}

extern "C" void kernel_launch(void* const* d_in, const int* in_sizes, int n_in,
                              void* d_out, int out_size, void* d_ws, size_t ws_size,
                              hipStream_t stream) {
    (void)n_in; (void)out_size; (void)d_ws; (void)ws_size;

    const float* in  = (const float*)d_in[0];
    float*       out = (float*)d_out;

    long long total = (long long)in_sizes[0];   // B*L*3 flat f32 elements
    long long npts  = total / 3;                // number of (x,y,z) points
    long long nquad = npts / 4;                 // 4 points per thread

    const int block = 256;                      // 8 wave32s per block

    if (nquad > 0) {
        long long grid = (nquad + block - 1) / block;
        cil_kernel_vec4<<<(unsigned int)grid, block, 0, stream>>>(
            (const float4*)in, (float4*)out, (int)nquad);
    }

    long long done = nquad * 4;
    long long rem  = npts - done;
    if (rem > 0) {
        long long grid = (rem + block - 1) / block;
        cil_kernel_tail<<<(unsigned int)grid, block, 0, stream>>>(
            in, out, (int)done, (int)npts);
    }
}